// ConcatAttention_43679817400720
// MI455X (gfx1250) — compile-verified
//
#include <hip/hip_runtime.h>

typedef __attribute__((ext_vector_type(2))) float v2f;
typedef __attribute__((ext_vector_type(8))) float v8f;

#define T_DIM 256
#define B_DIM 8
#define S_DIM 256
#define D_DIM 512   // QD == KD == VD == 512

// ---------------- CDNA5 fast paths (guarded; fall back cleanly) ------------
#if __has_builtin(__builtin_amdgcn_tanhf)
__device__ __forceinline__ float fast_tanh(float x) { return __builtin_amdgcn_tanhf(x); }
#else
__device__ __forceinline__ float fast_tanh(float x) { return tanhf(x); }
#endif

#if __has_builtin(__builtin_amdgcn_global_load_async_to_lds_b128) && \
    __has_builtin(__builtin_amdgcn_s_wait_asynccnt)
#define HAVE_ASYNC_LDS 1
#else
#define HAVE_ASYNC_LDS 0
#endif

#if HAVE_ASYNC_LDS
// Builtin signature (from hipcc diagnostic): param 1 is a non-const
// global-AS pointer to int32x4 (gcc vector_size(16)); param 2 is the LDS side.
typedef int gv4i __attribute__((vector_size(16)));
typedef __attribute__((address_space(1))) gv4i* as1_v4i_ptr;
typedef __attribute__((address_space(3))) gv4i* as3_v4i_ptr;
#endif

// Copy nf4 float4 elements global->LDS, strided across 256 threads.
__device__ __forceinline__ void lds_copy_f4(float* dst, const float* __restrict__ src,
                                            int tid, int nf4) {
#if HAVE_ASYNC_LDS
    for (int i = tid; i < nf4; i += 256) {
        __builtin_amdgcn_global_load_async_to_lds_b128(
            (as1_v4i_ptr)(src + 4 * i),
            (as3_v4i_ptr)(dst + 4 * i),
            0, 0);
    }
#else
    const float4* s4 = (const float4*)src;
    float4* d4 = (float4*)dst;
    for (int i = tid; i < nf4; i += 256) d4[i] = s4[i];
#endif
}

__device__ __forceinline__ void wait_async_lds() {
#if HAVE_ASYNC_LDS
    __builtin_amdgcn_s_wait_asynccnt(0);
#endif
}

// ---------------------------------------------------------------------------
// Kernel A: both input projections as fp32 WMMA GEMMs.
//   z=0: qp[m=b*T+t][k] = sum_q query[t,b,q] * Wa[k, q]        (Wq = Wa[:, :512])
//   z=1: kp[m=b*S+s][h] = sum_k keys[b,s,k]  * Wa[h, 512 + k]  (Wk = Wa[:, 512:])
// One wave computes one 16x16 output tile with V_WMMA_F32_16X16X4_F32.
//   A fragment (16x4 fp32): lane L holds M = L&15, K = k0 + (L>>4)*2 + {0,1}
//   B fragment (4x16 fp32): lane L holds N = L&15, K = k0 + (L>>4)*2 + {0,1}
//   C/D (16x16 fp32): vgpr r: lanes 0-15 -> M=r, lanes 16-31 -> M=r+8
// ---------------------------------------------------------------------------
__global__ __launch_bounds__(256) void proj_kernel(
    const float* __restrict__ query,   // (T,B,512)
    const float* __restrict__ keys,    // (B,S,512)
    const float* __restrict__ Wa,      // (512,1024)
    float* __restrict__ qp,            // (B*T,512)
    float* __restrict__ kp)            // (B*S,512)
{
    const int z    = blockIdx.y;            // 0 = qp, 1 = kp
    const int wave = threadIdx.x >> 5;
    const int lane = threadIdx.x & 31;
    const int half = lane >> 4;
    const int l15  = lane & 15;

    const int tile  = blockIdx.x * 8 + wave;  // 4096 tiles per z
    const int mTile = tile >> 5;              // 128 tiles of M=2048
    const int nTile = tile & 31;              // 32 tiles of N=512

    const int mRow = mTile * 16 + l15;
    const float* arow;
    if (z == 0) {
        const int b = mRow >> 8;              // m = b*T + t
        const int t = mRow & 255;
        arow = query + ((size_t)t * B_DIM + b) * D_DIM;
    } else {
        arow = keys + (size_t)mRow * D_DIM;   // m = b*S + s, contiguous
    }
    // Wa row = output column index n, Wa col = (z ? 512 : 0) + K
    const float* bcol = Wa + (size_t)(nTile * 16 + l15) * (2 * D_DIM) + (z ? D_DIM : 0);

    v8f acc = {};
    #pragma unroll 8
    for (int k0 = 0; k0 < D_DIM; k0 += 4) {
        const int kk = k0 + half * 2;
        v2f a = *(const v2f*)(arow + kk);     // 8B aligned (kk even)
        v2f b = *(const v2f*)(bcol + kk);
        acc = __builtin_amdgcn_wmma_f32_16x16x4_f32(
            false, a, false, b, (short)0, acc, false, false);
    }

    float* dst = (z == 0) ? qp : kp;
    const int col = nTile * 16 + l15;
    #pragma unroll
    for (int r = 0; r < 8; ++r) {
        const int row = mTile * 16 + r + half * 8;
        dst[(size_t)row * D_DIM + col] = acc[r];
    }
}

// ---------------------------------------------------------------------------
// Kernel B: fused tanh-reduce + softmax + ctx GEMM.
// One workgroup per (b, 16-row t-tile). 256 threads = 8 waves.
// kp is staged in 16-row chunks, double-buffered with async global->LDS DMA.
// ---------------------------------------------------------------------------
__global__ __launch_bounds__(256) void attn_kernel(
    const float* __restrict__ values,  // (B,S,512)
    const float* __restrict__ va,      // (512)
    const float* __restrict__ qp,      // (B*T,512)
    const float* __restrict__ kp,      // (B*S,512)
    float* __restrict__ outScore,      // (B,T,S)
    float* __restrict__ outCtx)        // (B,T,512)
{
    __shared__ float qpT[16 * D_DIM];        // 32 KB
    __shared__ float kpC[2][16 * D_DIM];     // 2 x 32 KB (double buffer)
    __shared__ float wbuf[16 * S_DIM];       // 16 KB: weights -> scores
    __shared__ float vaS[D_DIM];             //  2 KB

    const int tid = threadIdx.x;
    const int b   = blockIdx.y;
    const int t0  = blockIdx.x * 16;

    const int tl = tid >> 4;   // 0..15 : local t row
    const int sl = tid & 15;   // 0..15 : s within chunk

    // ---- issue initial async stages: qp tile, va, kp chunk 0 ----
    lds_copy_f4(qpT, qp + (size_t)(b * T_DIM + t0) * D_DIM, tid, 2048);
    lds_copy_f4(vaS, va, tid, 128);
    lds_copy_f4(kpC[0], kp + (size_t)(b * S_DIM) * D_DIM, tid, 2048);

    // ---- fused tanh-dot: weight[t, s] = sum_k va[k]*tanh(qp[t,k]+kp[s,k]) ----
    const int NCH = S_DIM / 16;              // 16 chunks of 16 s-rows
    for (int c = 0; c < NCH; ++c) {
        wait_async_lds();                    // this wave's outstanding DMAs done
        __syncthreads();                     // chunk c visible, prev readers done
        if (c + 1 < NCH)                     // prefetch next chunk into other slot
            lds_copy_f4(kpC[(c + 1) & 1],
                        kp + (size_t)(b * S_DIM + (c + 1) * 16) * D_DIM, tid, 2048);

        const float4* qrow = (const float4*)(qpT + tl * D_DIM);
        const float4* krow = (const float4*)(kpC[c & 1] + sl * D_DIM);
        const float4* va4  = (const float4*)vaS;
        float acc = 0.f;
        for (int k = 0; k < D_DIM / 4; ++k) {
            const float4 q = qrow[k], kk = krow[k], vv = va4[k];
            acc += vv.x * fast_tanh(q.x + kk.x);
            acc += vv.y * fast_tanh(q.y + kk.y);
            acc += vv.z * fast_tanh(q.z + kk.z);
            acc += vv.w * fast_tanh(q.w + kk.w);
        }
        wbuf[tl * S_DIM + c * 16 + sl] = acc;
    }
    __syncthreads();

    // ---- softmax over s (16 threads per row; xor-shuffles stay in 16-lane halves) ----
    {
        float* wr = wbuf + tl * S_DIM;
        float m = -3.402823466e38f;
        #pragma unroll
        for (int i = 0; i < 16; ++i) m = fmaxf(m, wr[sl * 16 + i]);
        #pragma unroll
        for (int x = 1; x < 16; x <<= 1) m = fmaxf(m, __shfl_xor(m, x, 32));
        float e[16];
        float sum = 0.f;
        #pragma unroll
        for (int i = 0; i < 16; ++i) { e[i] = __expf(wr[sl * 16 + i] - m); sum += e[i]; }
        #pragma unroll
        for (int x = 1; x < 16; x <<= 1) sum += __shfl_xor(sum, x, 32);
        const float inv = 1.0f / sum;
        float* so = outScore + (size_t)(b * T_DIM + t0 + tl) * S_DIM;
        #pragma unroll
        for (int i = 0; i < 16; ++i) {
            const float s = e[i] * inv;
            wr[sl * 16 + i] = s;
            so[sl * 16 + i] = s;
        }
    }
    __syncthreads();

    // ---- ctx = score(16x256) @ values[b](256x512) via WMMA f32 16x16x4 ----
    const int wave = tid >> 5;
    const int lane = tid & 31;
    const int half = lane >> 4;
    const int l15  = lane & 15;
    const float* arow = wbuf + l15 * S_DIM;                       // A: M = l15
    float* co = outCtx + ((size_t)b * T_DIM + t0) * D_DIM;

    for (int nt = wave; nt < 32; nt += 8) {
        const int n = nt * 16 + l15;
        const float* bcol = values + (size_t)b * S_DIM * D_DIM + n;
        v8f acc = {};
        #pragma unroll 4
        for (int k0 = 0; k0 < S_DIM; k0 += 4) {
            const int kk = k0 + half * 2;
            v2f a = *(const v2f*)(arow + kk);                     // LDS, 8B aligned
            v2f bb;
            bb.x = bcol[(size_t)kk * D_DIM];
            bb.y = bcol[(size_t)(kk + 1) * D_DIM];
            acc = __builtin_amdgcn_wmma_f32_16x16x4_f32(
                false, a, false, bb, (short)0, acc, false, false);
        }
        #pragma unroll
        for (int r = 0; r < 8; ++r) {
            const int row = r + half * 8;
            co[(size_t)row * D_DIM + n] = acc[r];
        }
    }
}

// ---------------------------------------------------------------------------
extern "C" void kernel_launch(void* const* d_in, const int* in_sizes, int n_in,
                              void* d_out, int out_size, void* d_ws, size_t ws_size,
                              hipStream_t stream) {
    const float* query  = (const float*)d_in[0];  // (256,8,512)
    const float* keys   = (const float*)d_in[1];  // (8,256,512)
    const float* values = (const float*)d_in[2];  // (8,256,512)
    const float* Wa     = (const float*)d_in[3];  // (512,1024)
    const float* va     = (const float*)d_in[4];  // (1,512)

    float* out      = (float*)d_out;
    float* outScore = out;                                  // B*T*S = 524288
    float* outCtx   = out + (size_t)B_DIM * T_DIM * S_DIM;  // B*T*512

    float* qp = (float*)d_ws;                   // 2048*512 floats = 4 MB
    float* kp = qp + (size_t)2048 * D_DIM;      // next 4 MB

    proj_kernel<<<dim3(512, 2), 256, 0, stream>>>(query, keys, Wa, qp, kp);
    attn_kernel<<<dim3(T_DIM / 16, B_DIM), 256, 0, stream>>>(values, va, qp, kp,
                                                             outScore, outCtx);
}